// DeepSetTM_86887188398184
// MI455X (gfx1250) — compile-verified
//
#include <hip/hip_runtime.h>

typedef __attribute__((ext_vector_type(2))) float v2f;
typedef __attribute__((ext_vector_type(4))) float v4f;
typedef __attribute__((ext_vector_type(8))) float v8f;
typedef __attribute__((ext_vector_type(4))) int   v4i;

#define N_ROWS 32768
#define DIN    128
#define DH     512
#define NOUT   10
#define FTRIM  512
#define TM     32      // rows of H per block in kernel 1
#define XS     132     // x-tile LDS stride   (132 % 64 == 4 -> conflict-free A-frag reads)
#define AS     516     // A-tile LDS stride   (516 % 64 == 4)
#define PS     528     // W-panel LDS stride  (528 % 64 == 16 -> half-waves hit disjoint banks)
#define KP     32      // K-panel depth

// ---- CDNA5 async global->LDS copy (ASYNCcnt-tracked), with safe fallback ----
#if defined(__has_builtin)
#  if __has_builtin(__builtin_amdgcn_global_load_async_to_lds_b128) && \
      __has_builtin(__builtin_amdgcn_s_wait_asynccnt)
#    define ASYNC_OK 1
#  endif
#endif

typedef __attribute__((address_space(1))) v4i g_v4i;   // global-memory int4
typedef __attribute__((address_space(3))) v4i l_v4i;   // LDS int4

__device__ __forceinline__ void async_cp16(const float* g, float* l) {
#ifdef ASYNC_OK
  __builtin_amdgcn_global_load_async_to_lds_b128((g_v4i*)g, (l_v4i*)l, 0, 0);
#else
  *(v4f*)l = *(const v4f*)g;
#endif
}
__device__ __forceinline__ void async_wait0() {
#ifdef ASYNC_OK
  __builtin_amdgcn_s_wait_asynccnt(0);
#endif
}

// D = A(16x4 f32) * B(4x16 f32) + C(16x16 f32), full-precision matrix pipe.
__device__ __forceinline__ v8f wmma_k4(v2f a, v2f b, v8f c) {
  return __builtin_amdgcn_wmma_f32_16x16x4_f32(false, a, false, b, (short)0, c, false, false);
}

// -------------------- Kernel 1: H^T = (relu(x@W1+b1)@W2+b2)^T --------------------
// Block: 256 threads = 8 wave32s; 32 rows of H, all 512 cols.
// Wave w: m-tile (w&1), n-group (w>>1) covering 8 n-tiles; 4 n-tiles share one A frag.
__global__ void __launch_bounds__(256) mlp_ht_kernel(
    const float* __restrict__ x,  const float* __restrict__ W1, const float* __restrict__ b1,
    const float* __restrict__ W2, const float* __restrict__ b2, float* __restrict__ HT)
{
  extern __shared__ float smem[];
  float* xs    = smem;                        // TM x XS   (~16.9 KB)
  float* at    = smem + TM * XS;              // TM x AS   (~66 KB)
  float* panel = smem + TM * XS + TM * AS;    // KP x PS   (~67.6 KB)

  const int tid  = threadIdx.x;
  const int lane = tid & 31;
  const int w    = tid >> 5;
  const int r0   = blockIdx.x * TM;
  const int half = lane >> 4;          // 0: K{0,1}/rows +0..7 ; 1: K{2,3}/rows +8..15
  const int l15  = lane & 15;
  const int mt   = w & 1;              // my m tile
  const int g    = w >> 1;             // my n-group (8 n-tiles)

  // stage x tile (32 x 128) into LDS via async copies (1 float4 per thread per iter)
  for (int it = 0; it < 4; ++it) {
    int idx = it * 256 + tid;          // 1024 float4s
    int row = idx >> 5;                // 32 float4 per row
    int c4  = (idx & 31) << 2;
    async_cp16(x + (size_t)(r0 + row) * DIN + c4, xs + row * XS + c4);
  }
  async_wait0();
  __syncthreads();

  v8f acc[8];

  // ---- Phase 1: A = relu(x @ W1 + b1) into LDS ----
#pragma unroll
  for (int u = 0; u < 8; ++u) acc[u] = (v8f){};
  for (int kp = 0; kp < DIN; kp += KP) {
    // stage W1 K-panel (32 x 512) into LDS
    for (int it = 0; it < 16; ++it) {
      int idx = it * 256 + tid;        // 4096 float4s
      int row = idx >> 7;              // 128 float4 per row
      int c4  = (idx & 127) << 2;
      async_cp16(W1 + (size_t)(kp + row) * DH + c4, panel + row * PS + c4);
    }
    async_wait0();
    __syncthreads();

    const float* ap0 = xs + (mt * 16 + l15) * XS + kp + half * 2;
    for (int q = 0; q < 2; ++q) {              // two groups of 4 concurrent n-tiles
#pragma unroll
      for (int kk = 0; kk < KP; kk += 4) {
        v2f a; a.x = ap0[kk]; a.y = ap0[kk + 1];
        const float* bp = panel + (kk + half * 2) * PS + (g * 8 + q * 4) * 16 + l15;
#pragma unroll
        for (int u = 0; u < 4; ++u) {
          v2f b; b.x = bp[u * 16]; b.y = bp[PS + u * 16];
          acc[q * 4 + u] = wmma_k4(a, b, acc[q * 4 + u]);
        }
      }
    }
    __syncthreads();
  }
  {
    const int mrow = mt * 16 + half * 8;
#pragma unroll
    for (int u = 0; u < 8; ++u) {
      int n = (g * 8 + u) * 16 + l15;
      float bias = b1[n];
#pragma unroll
      for (int i = 0; i < 8; ++i) {
        float v = acc[u][i] + bias;
        at[(mrow + i) * AS + n] = v > 0.f ? v : 0.f;
      }
    }
  }
  __syncthreads();

  // ---- Phase 2: H = A @ W2 + b2, stored transposed: HT[n * N_ROWS + row] ----
#pragma unroll
  for (int u = 0; u < 8; ++u) acc[u] = (v8f){};
  for (int kp = 0; kp < DH; kp += KP) {
    for (int it = 0; it < 16; ++it) {
      int idx = it * 256 + tid;
      int row = idx >> 7;
      int c4  = (idx & 127) << 2;
      async_cp16(W2 + (size_t)(kp + row) * DH + c4, panel + row * PS + c4);
    }
    async_wait0();
    __syncthreads();

    const float* ap0 = at + (mt * 16 + l15) * AS + kp + half * 2;
    for (int q = 0; q < 2; ++q) {
#pragma unroll
      for (int kk = 0; kk < KP; kk += 4) {
        v2f a; a.x = ap0[kk]; a.y = ap0[kk + 1];
        const float* bp = panel + (kk + half * 2) * PS + (g * 8 + q * 4) * 16 + l15;
#pragma unroll
        for (int u = 0; u < 4; ++u) {
          v2f b; b.x = bp[u * 16]; b.y = bp[PS + u * 16];
          acc[q * 4 + u] = wmma_k4(a, b, acc[q * 4 + u]);
        }
      }
    }
    __syncthreads();
  }
  {
    const int rowbase = r0 + mt * 16 + half * 8;    // multiple of 8 -> 16B aligned stores
#pragma unroll
    for (int u = 0; u < 8; ++u) {
      int n = (g * 8 + u) * 16 + l15;
      float bias = b2[n];
      v8f c = acc[u];
      v4f lo = { c[0] + bias, c[1] + bias, c[2] + bias, c[3] + bias };
      v4f hi = { c[4] + bias, c[5] + bias, c[6] + bias, c[7] + bias };
      float* dst = HT + (size_t)n * N_ROWS + rowbase;
      *(v4f*)(dst)     = lo;
      *(v4f*)(dst + 4) = hi;
    }
  }
}

// -------------------- Kernel 2: per-column trimmed mean via radix select --------------------
__device__ __forceinline__ unsigned f2key(float f) {
  unsigned u = __float_as_uint(f);
  return u ^ (unsigned)((((int)u) >> 31) | 0x80000000);   // monotone order-preserving map
}
__device__ __forceinline__ float key2f(unsigned k) {
  unsigned u = (k & 0x80000000u) ? (k ^ 0x80000000u) : ~k;
  return __uint_as_float(u);
}

__device__ unsigned radix_select(const float* col, unsigned* hist, unsigned* ctrl,
                                 int rank, int tid)
{
  unsigned prefix = 0, mask = 0;
  for (int shift = 24; shift >= 0; shift -= 8) {
    if (tid < 256) hist[tid] = 0;
    __syncthreads();
    for (int i = tid; i < N_ROWS; i += 256) {
      unsigned k = f2key(col[i]);
      if ((k & mask) == prefix) atomicAdd(&hist[(k >> shift) & 0xFF], 1u);
    }
    __syncthreads();
    if (tid == 0) {
      int run = 0; unsigned b = 0;
      for (; b < 255u; ++b) {
        int cnt = (int)hist[b];
        if (run + cnt > rank) break;
        run += cnt;
      }
      ctrl[0] = b;
      ctrl[1] = (unsigned)(rank - run);
    }
    __syncthreads();
    prefix |= ctrl[0] << shift;
    rank    = (int)ctrl[1];
    mask   |= 0xFFu << shift;
    __syncthreads();
  }
  return prefix;   // exact key of the rank-th element (bijective with float bits)
}

__global__ void __launch_bounds__(256) trimmed_mean_kernel(
    const float* __restrict__ HT, float* __restrict__ hbar)
{
  extern __shared__ float smem[];
  float*    col  = smem;                                   // 32768 floats (128 KB)
  unsigned* hist = (unsigned*)(smem + N_ROWS);             // 256
  unsigned* ctrl = (unsigned*)(smem + N_ROWS + 256);       // 2
  float*    fred = smem + N_ROWS + 258;                    // 3: total, sumLo, sumHi
  unsigned* ired = (unsigned*)(smem + N_ROWS + 261);       // 2: cntLo, cntHi

  const int tid = threadIdx.x;
  const int c   = blockIdx.x;
  const float* src = HT + (size_t)c * N_ROWS;

  // async-stage the whole column (contiguous thanks to the transposed H layout)
  for (int it = 0; it < 32; ++it) {
    int idx = (it * 256 + tid) << 2;     // float4 granularity
    async_cp16(src + idx, col + idx);
  }
  if (tid == 0) { fred[0] = fred[1] = fred[2] = 0.f; ired[0] = ired[1] = 0u; }
  async_wait0();
  __syncthreads();

  unsigned keyLo = radix_select(col, hist, ctrl, FTRIM - 1,      tid); // rank 511
  unsigned keyHi = radix_select(col, hist, ctrl, N_ROWS - FTRIM, tid); // rank 32256

  float tot = 0.f, slo = 0.f, shi = 0.f;
  unsigned clo = 0, chi = 0;
  for (int i = tid; i < N_ROWS; i += 256) {
    float v = col[i];
    unsigned k = f2key(v);
    tot += v;
    if (k < keyLo) { slo += v; ++clo; }
    if (k > keyHi) { shi += v; ++chi; }
  }
  atomicAdd(&fred[0], tot);
  atomicAdd(&fred[1], slo);
  atomicAdd(&fred[2], shi);
  atomicAdd(&ired[0], clo);
  atomicAdd(&ired[1], chi);
  __syncthreads();

  if (tid == 0) {
    // trim exactly FTRIM from each end (exact tie handling)
    float trim = fred[1] + (float)(FTRIM - (int)ired[0]) * key2f(keyLo)
               + fred[2] + (float)(FTRIM - (int)ired[1]) * key2f(keyHi);
    hbar[c] = (fred[0] - trim) / (float)(N_ROWS - 2 * FTRIM);
  }
}

// -------------------- Kernel 3: logits = relu(hbar@W3+b3)@W4+b4 --------------------
__global__ void __launch_bounds__(512) decode_kernel(
    const float* __restrict__ hbar, const float* __restrict__ W3, const float* __restrict__ b3,
    const float* __restrict__ W4,   const float* __restrict__ b4, float* __restrict__ out)
{
  __shared__ float hb[DH];
  __shared__ float g[DH];
  const int tid = threadIdx.x;
  hb[tid] = hbar[tid];
  __syncthreads();
  float acc = b3[tid];
  for (int k = 0; k < DH; ++k) acc += hb[k] * W3[(size_t)k * DH + tid];
  g[tid] = acc > 0.f ? acc : 0.f;
  __syncthreads();
  if (tid < NOUT) {
    float o = b4[tid];
    for (int k = 0; k < DH; ++k) o += g[k] * W4[(size_t)k * NOUT + tid];
    out[tid] = o;
  }
}

// -------------------- launcher --------------------
extern "C" void kernel_launch(void* const* d_in, const int* in_sizes, int n_in,
                              void* d_out, int out_size, void* d_ws, size_t ws_size,
                              hipStream_t stream) {
  (void)in_sizes; (void)n_in; (void)out_size; (void)ws_size;
  const float* x  = (const float*)d_in[0];
  const float* W1 = (const float*)d_in[1];
  const float* b1 = (const float*)d_in[2];
  const float* W2 = (const float*)d_in[3];
  const float* b2 = (const float*)d_in[4];
  const float* W3 = (const float*)d_in[5];
  const float* b3 = (const float*)d_in[6];
  const float* W4 = (const float*)d_in[7];
  const float* b4 = (const float*)d_in[8];
  // d_in[9] = f (== 512), baked in as FTRIM

  float* HT   = (float*)d_ws;                     // 512 * 32768 floats = 64 MB
  float* hbar = HT + (size_t)DH * N_ROWS;         // 512 floats

  size_t lds1 = (size_t)(TM * XS + TM * AS + KP * PS) * sizeof(float);  // ~150.5 KB
  mlp_ht_kernel<<<N_ROWS / TM, 256, lds1, stream>>>(x, W1, b1, W2, b2, HT);

  size_t lds2 = (size_t)(N_ROWS + 272) * sizeof(float);                 // ~129 KB
  trimmed_mean_kernel<<<DH, 256, lds2, stream>>>(HT, hbar);

  decode_kernel<<<1, DH, 0, stream>>>(hbar, W3, b3, W4, b4, (float*)d_out);
}